// optimNet_16475494548224
// MI455X (gfx1250) — compile-verified
//
#include <hip/hip_runtime.h>
#include <stdint.h>

// ---------------------------------------------------------------------------
// 2-layer GCN for MI455X (gfx1250, wave32).
// GEMMs: V_WMMA_F32_16X16X4_F32, 32x128 block tile, A/B tiles staged into
// LDS by the Tensor Data Mover (TENSOR_LOAD_TO_LDS, ping-pong buffers,
// s_wait_tensorcnt) -- register-double-buffer fallback if the TDM builtin is
// absent. Aggregation: L2-resident coalesced float4 gather + atomic_add_f32.
// ---------------------------------------------------------------------------

typedef float v2f __attribute__((ext_vector_type(2)));
typedef float v8f __attribute__((ext_vector_type(8)));
typedef unsigned int uint32x4 __attribute__((ext_vector_type(4)));
typedef int          int32x8  __attribute__((ext_vector_type(8)));
typedef int          int32x4  __attribute__((ext_vector_type(4)));

#if defined(__has_builtin)
#if __has_builtin(__builtin_amdgcn_tensor_load_to_lds)
#define USE_TDM 1
#endif
#endif
#ifndef USE_TDM
#define USE_TDM 0
#endif

#define BM 32
#define BN 128
#define KC 32

#if USE_TDM
// --------------------------------------------------------------------------
// TDM 2-D tile load: tile_d1 rows x tile_d0 elems (f32) from row-major
// tensor (row pitch stride0 elems) at gaddr -> LDS at lds_off, inserting
// pad_amount+1 DWORDs of padding every 2<<pad_interval DWORDs (ISA 8.4).
// Descriptor layout per CDNA5 ISA 8.3/8.4; trailing groups unused for 2-D.
// This toolchain exposes the 6-arg builtin:
//   (uint32x4 g0, int32x8 g1, int32x4, int32x4, int32x8, i32 cpol)
// --------------------------------------------------------------------------
__device__ static inline void tdm_load_2d(
    const float* gaddr, unsigned lds_off,
    unsigned tensor_d0, unsigned tensor_d1,
    unsigned tile_d0, unsigned tile_d1,
    unsigned long long stride0, unsigned pad_interval, unsigned pad_amount)
{
    unsigned long long ga = (unsigned long long)(uintptr_t)gaddr;
    uint32x4 g0;
    g0[0] = 1u;                                   // count=1 (valid), user mode
    g0[1] = lds_off;                              // lds_addr (bytes)
    g0[2] = (unsigned)(ga & 0xffffffffu);         // global_addr[31:0]
    g0[3] = (unsigned)((ga >> 32) & 0x01ffffffu)  // global_addr[56:32]
          | (2u << 30);                           // type = 2 ("image")
    int32x8 g1;
    unsigned w0 = (2u << 16);                     // data_size = 2 -> 4 bytes
    if (pad_amount)
        w0 |= (1u << 20) | (pad_interval << 22) | (pad_amount << 25);
    g1[0] = (int)w0;
    g1[1] = (int)((tensor_d0 & 0xffffu) << 16);               // dim0[15:0]
    g1[2] = (int)(((tensor_d0 >> 16) & 0xffffu)
          | ((tensor_d1 & 0xffffu) << 16));                   // dim0[31:16]|dim1[15:0]
    g1[3] = (int)(((tensor_d1 >> 16) & 0xffffu)
          | ((tile_d0 & 0xffffu) << 16));                     // dim1[31:16]|tile_d0
    g1[4] = (int)(tile_d1 & 0xffffu);                         // tile_d1; tile_d2=0
    g1[5] = (int)(unsigned)(stride0 & 0xffffffffu);           // stride0[31:0]
    g1[6] = (int)(unsigned)((stride0 >> 32) & 0xffffu);       // stride0[47:32]
    g1[7] = 0;                                                // dim1_stride unused (2-D)
    int32x4 z4 = {0, 0, 0, 0};
    int32x8 z8 = {0, 0, 0, 0, 0, 0, 0, 0};
    __builtin_amdgcn_tensor_load_to_lds(g0, g1, z4, z4, z8, 0);
}
#endif

// ---------------------------------------------------------------------------
// C[M,N] = A[M,K] @ B[K,N], row-major f32.  256 threads = 8 waves.
// Wave grid 2(M) x 4(N); each wave computes 16 rows x 32 cols (2 acc tiles).
// Requires N % BN == 0, K % KC == 0 (true here: N in {256,128}, K in {512,256}).
// ---------------------------------------------------------------------------
__global__ __launch_bounds__(256) void gemm_wmma(
    const float* __restrict__ A, const float* __restrict__ B,
    float* __restrict__ C, int M, int N, int K)
{
#if USE_TDM
    __shared__ __align__(16) float As[2][BM][KC + 4];   // pitch 36 fl (pad 4/32)
    __shared__ __align__(16) float Bs[2][KC][BN + 4];   // pitch 132 fl (pad 4/128)
#else
    __shared__ __align__(16) float As[1][BM][KC + 4];
    __shared__ __align__(16) float Bs[1][KC][BN + 4];
#endif

    const int tid  = threadIdx.x;
    const int lane = tid & 31;
    const int wid  = tid >> 5;
    const int g    = lane >> 4;        // half-wave group (0/1)
    const int r    = lane & 15;

    const int bm = blockIdx.x * BM;
    const int bn = blockIdx.y * BN;
    const int wm = (wid & 1) * 16;     // wave row offset
    const int wn = (wid >> 1) * 32;    // wave col offset (2 tiles: wn, wn+16)

    int p = 0;

#if USE_TDM
    // Scalar-uniform wave id: tensor ops IGNORE EXEC, so the issue must sit
    // behind a true scalar branch or every wave would duplicate the DMA.
    const int uw = __builtin_amdgcn_readfirstlane(wid);
    if (uw == 0)
        tdm_load_2d(A + (size_t)bm * K, (unsigned)(uintptr_t)&As[0][0][0],
                    (unsigned)K, (unsigned)(M - bm), KC, BM,
                    (unsigned long long)K, 4u, 3u);      // pad 4 every 32 dw
    if (uw == 1)
        tdm_load_2d(B + bn, (unsigned)(uintptr_t)&Bs[0][0][0],
                    (unsigned)(N - bn), (unsigned)K, BN, KC,
                    (unsigned long long)N, 6u, 3u);      // pad 4 every 128 dw
    __builtin_amdgcn_s_wait_tensorcnt(0);                // no-op for non-issuers
    __syncthreads();
#else
    // fallback: float4 global staging, register double buffer
    const int arow = tid >> 3;
    const int acol = (tid & 7) * 4;
    int ag = bm + arow; if (ag >= M) ag = M - 1;         // stores are guarded
    const float* aptr = A + (size_t)ag * K + acol;
    const int brow = tid >> 5;
    const int bcol = (tid & 31) * 4;
    const float* bptr = B + (size_t)brow * N + bn + bcol;
    float4 aR = *(const float4*)aptr;
    float4 bR[4];
    #pragma unroll
    for (int i = 0; i < 4; ++i) bR[i] = *(const float4*)(bptr + (size_t)(i * 8) * N);
#endif

    v8f acc0 = {}, acc1 = {};

    for (int k0 = 0; k0 < K; k0 += KC) {
#if USE_TDM
        // kick next chunk's DMA into the other buffer (hidden by WMMAs)
        if (k0 + KC < K) {
            const int kn = k0 + KC;
            if (uw == 0)
                tdm_load_2d(A + (size_t)bm * K + kn,
                            (unsigned)(uintptr_t)&As[p ^ 1][0][0],
                            (unsigned)(K - kn), (unsigned)(M - bm), KC, BM,
                            (unsigned long long)K, 4u, 3u);
            if (uw == 1)
                tdm_load_2d(B + (size_t)kn * N + bn,
                            (unsigned)(uintptr_t)&Bs[p ^ 1][0][0],
                            (unsigned)(N - bn), (unsigned)(K - kn), BN, KC,
                            (unsigned long long)N, 6u, 3u);
        }
#else
        *(float4*)&As[0][arow][acol] = aR;
        #pragma unroll
        for (int i = 0; i < 4; ++i) *(float4*)&Bs[0][brow + i * 8][bcol] = bR[i];
        __syncthreads();
        if (k0 + KC < K) {
            aptr += KC;
            bptr += (size_t)KC * N;
            aR = *(const float4*)aptr;
            #pragma unroll
            for (int i = 0; i < 4; ++i) bR[i] = *(const float4*)(bptr + (size_t)(i * 8) * N);
        }
#endif

        // preload all 8 k-step fragments, then 16 back-to-back WMMAs
        v2f af[8], bf0[8], bf1[8];
        #pragma unroll
        for (int s = 0; s < 8; ++s) {
            const int kk = s * 4 + g * 2;   // A 16x4: lanes 0-15 K{0,1}, 16-31 K{2,3}
            af[s].x  = As[p][wm + r][kk];      af[s].y  = As[p][wm + r][kk + 1];
            bf0[s].x = Bs[p][kk][wn + r];      bf0[s].y = Bs[p][kk + 1][wn + r];
            bf1[s].x = Bs[p][kk][wn + 16 + r]; bf1[s].y = Bs[p][kk + 1][wn + 16 + r];
        }
        #pragma unroll
        for (int s = 0; s < 8; ++s) {
            acc0 = __builtin_amdgcn_wmma_f32_16x16x4_f32(
                false, af[s], false, bf0[s], (short)0, acc0, false, false);
            acc1 = __builtin_amdgcn_wmma_f32_16x16x4_f32(
                false, af[s], false, bf1[s], (short)0, acc1, false, false);
        }

#if USE_TDM
        __builtin_amdgcn_s_wait_tensorcnt(0);   // issuer: DMA landed in LDS
        __syncthreads();                        // everyone: buffer handoff
        p ^= 1;
#else
        __syncthreads();
#endif
    }

    // C/D 16x16 f32 layout: VGPR v, lane(g,r) -> row = v + 8*g, col = r.
    #pragma unroll
    for (int v = 0; v < 8; ++v) {
        int row = bm + wm + g * 8 + v;
        if (row < M) {
            float* cp = C + (size_t)row * N + bn + wn + r;
            cp[0]  = acc0[v];
            cp[16] = acc1[v];
        }
    }
}

// ---------------------------------------------------------------------------
// Small per-node / per-edge kernels
// ---------------------------------------------------------------------------
__global__ void fill_f32(float* __restrict__ p, float v, int n)
{
    int i = blockIdx.x * blockDim.x + threadIdx.x;
    if (i < n) p[i] = v;
}

// deg[dst] += ew[e]   (deg pre-initialized to 1.0 == self-loop weight)
__global__ void edge_deg(const long long* __restrict__ ei,
                         const float* __restrict__ w,
                         float* __restrict__ deg, int nE)
{
    int e = blockIdx.x * blockDim.x + threadIdx.x;
    if (e < nE) atomicAdd(deg + ei[nE + e], w[e]);
}

// in-place deg -> deg^{-1/2}
__global__ void deg_to_dinv(float* __restrict__ deg, int n)
{
    int i = blockIdx.x * blockDim.x + threadIdx.x;
    if (i < n) {
        float d = deg[i];
        deg[i] = (d > 0.f) ? rsqrtf(d) : 0.f;
    }
}

// norm[e] = dinv[src] * w[e] * dinv[dst]
__global__ void edge_norm(const long long* __restrict__ ei,
                          const float* __restrict__ w,
                          const float* __restrict__ dinv,
                          float* __restrict__ nrm, int nE)
{
    int e = blockIdx.x * blockDim.x + threadIdx.x;
    if (e < nE) nrm[e] = dinv[ei[e]] * w[e] * dinv[ei[nE + e]];
}

// out[dst,:] += norm[e] * h[src,:]  -- feature index fastest => coalesced.
// float4 gather + 4 atomics per thread; node tables are L2-resident.
__global__ void edge_scatter(const long long* __restrict__ ei,
                             const float* __restrict__ nrm,
                             const float* __restrict__ h,
                             float* __restrict__ out, int nE, int F)
{
    const int chunk = F >> 2;
    long long idx = (long long)blockIdx.x * blockDim.x + threadIdx.x;
    long long tot = (long long)nE * chunk;
    if (idx >= tot) return;
    int e  = (int)(idx / chunk);
    int f4 = (int)(idx - (long long)e * chunk) << 2;
    long long s0 = ei[e];
    long long d0 = ei[nE + e];
    float nm = nrm[e];
    const float4 hv = *(const float4*)(h + s0 * F + f4);
    float* op = out + d0 * F + f4;
    atomicAdd(op + 0, hv.x * nm);
    atomicAdd(op + 1, hv.y * nm);
    atomicAdd(op + 2, hv.z * nm);
    atomicAdd(op + 3, hv.w * nm);
}

// out[i,f] = (acc + h[i,f]*dinv[i]^2 + bias[f]) , optional relu (self-loop
// term folded in here -> no atomics needed for the n self-loop edges)
__global__ void finalize_node(const float* __restrict__ h,
                              const float* __restrict__ dinv,
                              const float* __restrict__ bias,
                              float* __restrict__ out,
                              int n, int F, int do_relu)
{
    long long idx = (long long)blockIdx.x * blockDim.x + threadIdx.x;
    if (idx >= (long long)n * F) return;
    int i = (int)(idx / F);
    int f = (int)(idx - (long long)i * F);
    float d = dinv[i];
    float v = out[idx] + h[idx] * d * d + bias[f];
    out[idx] = do_relu ? fmaxf(v, 0.f) : v;
}

// s[i] = out1[i,:] . Wm[0:F] ; t[i] = out1[i,:] . Wm[F:2F]   (wave per node)
__global__ __launch_bounds__(256) void node_dot(
    const float* __restrict__ xr, const float* __restrict__ Wm,
    float* __restrict__ s, float* __restrict__ t, int n, int F)
{
    int w    = blockIdx.x * (blockDim.x >> 5) + (threadIdx.x >> 5);
    int lane = threadIdx.x & 31;
    if (w >= n) return;
    const float* row = xr + (size_t)w * F;
    float ss = 0.f, tt = 0.f;
    for (int j = lane; j < F; j += 32) {
        float xv = row[j];
        ss = fmaf(xv, Wm[j], ss);
        tt = fmaf(xv, Wm[F + j], tt);
    }
    #pragma unroll
    for (int off = 16; off; off >>= 1) {
        ss += __shfl_xor(ss, off, 32);
        tt += __shfl_xor(tt, off, 32);
    }
    if (lane == 0) { s[w] = ss; t[w] = tt; }
}

// e_attr[e] = relu(s[src] + t[dst] + bm)
__global__ void edge_mlp(const long long* __restrict__ ei,
                         const float* __restrict__ s,
                         const float* __restrict__ t,
                         const float* __restrict__ bm,
                         float* __restrict__ ea, int nE)
{
    int e = blockIdx.x * blockDim.x + threadIdx.x;
    if (e < nE) ea[e] = fmaxf(s[ei[e]] + t[ei[nE + e]] + bm[0], 0.f);
}

// ---------------------------------------------------------------------------
extern "C" void kernel_launch(void* const* d_in, const int* in_sizes, int n_in,
                              void* d_out, int out_size, void* d_ws, size_t ws_size,
                              hipStream_t stream)
{
    const float*     x  = (const float*)d_in[0];      // 50000 x 512
    const long long* ei = (const long long*)d_in[1];  // 2 x nE (int64)
    const float*     ew = (const float*)d_in[2];      // nE
    const float*     W1 = (const float*)d_in[3];      // 512 x 256
    const float*     b1 = (const float*)d_in[4];      // 256
    const float*     W2 = (const float*)d_in[5];      // 256 x 128
    const float*     b2 = (const float*)d_in[6];      // 128
    const float*     Wm = (const float*)d_in[7];      // 512
    const float*     bm = (const float*)d_in[8];      // 1
    float* out = (float*)d_out;                       // 50000 x 128

    const int K1 = 512, F1 = 256, F2 = 128;
    const int nN = in_sizes[0] / K1;
    const int nE = in_sizes[1] / 2;

    // workspace carve-out (~110 MB)
    float* ws   = (float*)d_ws;
    float* h    = ws; ws += (size_t)nN * F1;   // h1, later reused as h2
    float* out1 = ws; ws += (size_t)nN * F1;
    float* deg  = ws; ws += nN;                // deg -> dinv, reused per layer
    float* sbuf = ws; ws += nN;
    float* tbuf = ws; ws += nN;
    float* nrm  = ws; ws += nE;
    float* ea   = ws; ws += nE;

    const int T  = 256;
    const int gN = (nN + T - 1) / T;
    const int gE = (nE + T - 1) / T;

    // ---- layer 1 normalization -------------------------------------------
    fill_f32<<<gN, T, 0, stream>>>(deg, 1.0f, nN);            // self-loop wt
    edge_deg<<<gE, T, 0, stream>>>(ei, ew, deg, nE);
    deg_to_dinv<<<gN, T, 0, stream>>>(deg, nN);
    edge_norm<<<gE, T, 0, stream>>>(ei, ew, deg, nrm, nE);

    // ---- h1 = X @ W1 (WMMA + TDM staging) --------------------------------
    {
        dim3 grid((nN + BM - 1) / BM, F1 / BN);
        gemm_wmma<<<grid, T, 0, stream>>>(x, W1, h, nN, F1, K1);
    }

    // ---- aggregate layer 1, add bias, relu -------------------------------
    fill_f32<<<(nN * F1 + T - 1) / T, T, 0, stream>>>(out1, 0.f, nN * F1);
    {
        long long tot = (long long)nE * (F1 >> 2);
        edge_scatter<<<(int)((tot + T - 1) / T), T, 0, stream>>>(ei, nrm, h, out1, nE, F1);
    }
    finalize_node<<<(nN * F1 + T - 1) / T, T, 0, stream>>>(h, deg, b1, out1, nN, F1, 1);

    // ---- edge MLP: e_attr = relu(out1[src].Wm_a + out1[dst].Wm_b + bm) ---
    node_dot<<<(nN * 32 + T - 1) / T, T, 0, stream>>>(out1, Wm, sbuf, tbuf, nN, F1);
    edge_mlp<<<gE, T, 0, stream>>>(ei, sbuf, tbuf, bm, ea, nE);

    // ---- layer 2 normalization (weights = e_attr) ------------------------
    fill_f32<<<gN, T, 0, stream>>>(deg, 1.0f, nN);
    edge_deg<<<gE, T, 0, stream>>>(ei, ea, deg, nE);
    deg_to_dinv<<<gN, T, 0, stream>>>(deg, nN);
    edge_norm<<<gE, T, 0, stream>>>(ei, ea, deg, nrm, nE);

    // ---- h2 = out1 @ W2 (WMMA), reuse h buffer ---------------------------
    {
        dim3 grid((nN + BM - 1) / BM, F2 / BN);
        gemm_wmma<<<grid, T, 0, stream>>>(out1, W2, h, nN, F2, F1);
    }

    // ---- aggregate layer 2 into d_out, add bias (no relu) ----------------
    fill_f32<<<(nN * F2 + T - 1) / T, T, 0, stream>>>(out, 0.f, nN * F2);
    {
        long long tot = (long long)nE * (F2 >> 2);
        edge_scatter<<<(int)((tot + T - 1) / T), T, 0, stream>>>(ei, nrm, h, out, nE, F2);
    }
    finalize_node<<<(nN * F2 + T - 1) / T, T, 0, stream>>>(h, deg, b2, out, nN, F2, 0);
}